// Net_71871982731665
// MI455X (gfx1250) — compile-verified
//
#include <hip/hip_runtime.h>
#include <math.h>

typedef float v2f __attribute__((ext_vector_type(2)));
typedef float v8f __attribute__((ext_vector_type(8)));

// workspace layout (floats)
#define WS_SUMSQ 0      // 1 float : global sum of squares of enc
#define WS_T     8      // 49 floats : 7x7 effective channel matrix
#define WS_ENC   64     // B*7 floats : pre-norm encoder output

#define SAMPS  20
#define KTAPS  101
#define SIGLEN 140      // CD(7) * SAMPS(20)
#define LEAK   0.01f
#define PI_F   3.14159265358979323846f

// ---------------------------------------------------------------------------
// Kernel A: build the 7x7 effective linear operator T of the truncated
// (same-padded) conv chain: hp -> hl -> hp on a length-140 signal, sampled
// at positions 20*j from impulses at 20*c.  One block per impulse position.
// Also zeroes the global sum-of-squares accumulator.
// ---------------------------------------------------------------------------
__global__ void tdm_filt_kernel(const float* __restrict__ h_ps,
                                const float* __restrict__ h_lp,
                                float* __restrict__ ws) {
  __shared__ float a[SIGLEN];
  __shared__ float b[SIGLEN];
  const int c = blockIdx.x;
  const int t = threadIdx.x;
  if (c == 0 && t == 0) ws[WS_SUMSQ] = 0.0f;
  if (t < SIGLEN) a[t] = (t == SAMPS * c) ? 1.0f : 0.0f;
  __syncthreads();
  if (t < SIGLEN) {                 // conv same with h_ps : a -> b
    float s = 0.f;
    for (int j = 0; j < KTAPS; ++j) { int q = t + 50 - j; if (q >= 0 && q < SIGLEN) s += h_ps[j] * a[q]; }
    b[t] = s;
  }
  __syncthreads();
  if (t < SIGLEN) {                 // conv same with h_lp : b -> a
    float s = 0.f;
    for (int j = 0; j < KTAPS; ++j) { int q = t + 50 - j; if (q >= 0 && q < SIGLEN) s += h_lp[j] * b[q]; }
    a[t] = s;
  }
  __syncthreads();
  if (t < SIGLEN) {                 // conv same with h_ps : a -> b
    float s = 0.f;
    for (int j = 0; j < KTAPS; ++j) { int q = t + 50 - j; if (q >= 0 && q < SIGLEN) s += h_ps[j] * a[q]; }
    b[t] = s;
  }
  __syncthreads();
  if (t < SIGLEN && (t % SAMPS) == 0) {
    const int j = t / SAMPS;
    ws[WS_T + j * 7 + c] = (float)SAMPS * b[t];   // y = sig[::20] * 20
  }
}

// ---------------------------------------------------------------------------
// Kernel B: encoder.  One wave per 16-row tile.  WMMA f32 16x16x4 chained
// over K.  A-frag: x rows (float2/lane direct from global).  B-frag: W^T
// (float2/lane direct from global).  h tile restaged via per-wave LDS.
// Writes pre-norm enc (B x 7) + accumulates global sum of squares.
// ---------------------------------------------------------------------------
__global__ __launch_bounds__(256) void enc_kernel(
    const float* __restrict__ x,
    const float* __restrict__ We1, const float* __restrict__ be1,
    const float* __restrict__ We2, const float* __restrict__ be2,
    float* __restrict__ ws, int nrows) {
  __shared__ float tile[8][256];
  const int lane = threadIdx.x & 31;
  const int wid  = threadIdx.x >> 5;
  const int rowBase = (blockIdx.x * 8 + wid) * 16;
  if (rowBase + 16 > nrows) return;
  const int m    = lane & 15;     // row within tile (A layout)
  const int half = lane >> 4;     // K-half selector for A/B fragments
  const int n    = m;             // output column (B/C layouts)

  // ---- layer 1: h = leaky(x @ We1^T + be1), K = 16 as 4 WMMA k-chunks ----
  v8f acc = {0.f,0.f,0.f,0.f,0.f,0.f,0.f,0.f};
#pragma unroll
  for (int kc = 0; kc < 4; ++kc) {
    const int k0 = kc * 4 + half * 2;
    v2f av = *(const v2f*)(x + (rowBase + m) * 16 + k0);     // A[m][k0..k0+1]
    v2f bv = *(const v2f*)(We1 + n * 16 + k0);               // B[k][n] = We1[n][k]
    acc = __builtin_amdgcn_wmma_f32_16x16x4_f32(false, av, false, bv,
                                                (short)0, acc, false, false);
  }
  {
    const float bias = be1[n];
#pragma unroll
    for (int i = 0; i < 8; ++i) {
      float v = acc[i] + bias;
      acc[i] = (v >= 0.f) ? v : LEAK * v;
    }
  }

  // restage h tile (C layout -> LDS row-major -> A fragments)
  float* L = &tile[wid][0];
#pragma unroll
  for (int i = 0; i < 8; ++i) L[(i + half * 8) * 16 + n] = acc[i];
  __builtin_amdgcn_wave_barrier();   // keep ds store/load order (in-order LDS)

  // ---- layer 2: enc = h @ We2^T + be2  (N=7 padded to 16) ----
  v8f acc2 = {0.f,0.f,0.f,0.f,0.f,0.f,0.f,0.f};
#pragma unroll
  for (int kc = 0; kc < 4; ++kc) {
    const int k0 = kc * 4 + half * 2;
    v2f av = { L[m * 16 + k0], L[m * 16 + k0 + 1] };
    v2f bv;
    if (n < 7) bv = *(const v2f*)(We2 + n * 16 + k0);
    else { bv.x = 0.f; bv.y = 0.f; }
    acc2 = __builtin_amdgcn_wmma_f32_16x16x4_f32(false, av, false, bv,
                                                 (short)0, acc2, false, false);
  }

  float local = 0.f;
  if (n < 7) {
    const float bias = be2[n];
    float* encp = ws + WS_ENC;
#pragma unroll
    for (int i = 0; i < 8; ++i) {
      const int mm = i + half * 8;
      float v = acc2[i] + bias;
      encp[(rowBase + mm) * 7 + n] = v;
      local += v * v;
    }
  }
#pragma unroll
  for (int off = 16; off > 0; off >>= 1) local += __shfl_xor(local, off, 32);
  if (lane == 0) atomicAdd(ws + WS_SUMSQ, local);
}

// ---------------------------------------------------------------------------
// Kernel C: normalize/quantize/channel/requantize + decoder (both decoder
// GEMMs via WMMA f32 16x16x4, staged through per-wave LDS).  The
// transcendental channel stage runs on all 32 lanes in two LDS-staged
// passes over the 16x7 = 112 symbols of the wave's tile.
// ---------------------------------------------------------------------------
__global__ __launch_bounds__(256) void dec_kernel(
    const float* __restrict__ ws,
    const float* __restrict__ Wd1, const float* __restrict__ bd1,
    const float* __restrict__ Wd2, const float* __restrict__ bd2,
    const float* __restrict__ noise, const int* __restrict__ snrp,
    float* __restrict__ out, int nrows) {
  __shared__ float dtile[8][256];   // 16x16 d tile per wave
  __shared__ float rtile[8][128];   // 16x8  res tile per wave (K padded 7->8)
  __shared__ float ctile[8][336];   // per wave: sr[112] | si[112] | sg[112]
  const int lane = threadIdx.x & 31;
  const int wid  = threadIdx.x >> 5;
  const int rowBase = (blockIdx.x * 8 + wid) * 16;
  if (rowBase + 16 > nrows) return;
  const int m    = lane & 15;
  const int half = lane >> 4;
  const int n    = m;

  const float scale   = 70.0f / sqrtf(ws[WS_SUMSQ]);
  const float snr_lin = powf(10.0f, 0.1f * (float)snrp[0]);
  const float nf      = 1.0f / sqrtf(2.0f * (4.0f / 7.0f) * snr_lin);
  const float* T    = ws + WS_T;
  const float* encp = ws + WS_ENC;

  float* R = &rtile[wid][0];
  float* C = &ctile[wid][0];

  // ---- pass 1: normalize -> round -> PSK phase (all 32 lanes) ----
  if (lane < 16) R[lane * 8 + 7] = 0.f;          // K padding column
  for (int t = lane; t < 112; t += 32) {         // t = mm*7 + c
    float e  = encp[rowBase * 7 + t] * scale;
    float xf = rintf(e);                         // RNE, matches jnp.round
    int   xi = (int)xf;
    int  idx = (xi < 0 ? -xi : xi) & 127;        // mod 2^CD
    float ph = (2.0f * PI_F / 128.0f) * (float)idx;
    C[t]       = cosf(ph);
    C[112 + t] = sinf(ph);
    C[224 + t] = (xf < 0.f) ? -1.f : 1.f;
  }
  __builtin_amdgcn_wave_barrier();

  // ---- pass 2: 7x7 channel matvec -> atan2 -> requantize (+noise) ----
  for (int t = lane; t < 112; t += 32) {         // t = mm*7 + j
    const int mm = t / 7;
    const int j  = t - mm * 7;
    float yr = 0.f, yi = 0.f;
#pragma unroll
    for (int c = 0; c < 7; ++c) {
      const float tv = T[j * 7 + c];
      yr += tv * C[mm * 7 + c];
      yi += tv * C[112 + mm * 7 + c];
    }
    float ang = atan2f(yi, yr);
    float r   = rintf(ang * (128.0f / (2.0f * PI_F)));
    if (r < 0.f) r += 128.0f;                    // mod 128, |r| <= 64
    R[mm * 8 + j] = r * C[224 + t] + noise[rowBase * 7 + t] * nf;
  }
  __builtin_amdgcn_wave_barrier();

  // ---- decoder layer 1: d = leaky(res @ Wd1^T + bd1), K = 8 (7 + pad) ----
  v8f acc = {0.f,0.f,0.f,0.f,0.f,0.f,0.f,0.f};
#pragma unroll
  for (int kc = 0; kc < 2; ++kc) {
    const int k0 = kc * 4 + half * 2;
    v2f av = { R[m * 8 + k0], R[m * 8 + k0 + 1] };
    v2f bv;
    bv.x = (k0     < 7) ? Wd1[n * 7 + k0]     : 0.f;   // B[k][n] = Wd1[n][k]
    bv.y = (k0 + 1 < 7) ? Wd1[n * 7 + k0 + 1] : 0.f;
    acc = __builtin_amdgcn_wmma_f32_16x16x4_f32(false, av, false, bv,
                                                (short)0, acc, false, false);
  }
  {
    const float bias = bd1[n];
#pragma unroll
    for (int i = 0; i < 8; ++i) {
      float v = acc[i] + bias;
      acc[i] = (v >= 0.f) ? v : LEAK * v;
    }
  }
  float* L = &dtile[wid][0];
#pragma unroll
  for (int i = 0; i < 8; ++i) L[(i + half * 8) * 16 + n] = acc[i];
  __builtin_amdgcn_wave_barrier();

  // ---- decoder layer 2: out = d @ Wd2^T + bd2, K = 16 ----
  v8f acc2 = {0.f,0.f,0.f,0.f,0.f,0.f,0.f,0.f};
#pragma unroll
  for (int kc = 0; kc < 4; ++kc) {
    const int k0 = kc * 4 + half * 2;
    v2f av = { L[m * 16 + k0], L[m * 16 + k0 + 1] };
    v2f bv = *(const v2f*)(Wd2 + n * 16 + k0);
    acc2 = __builtin_amdgcn_wmma_f32_16x16x4_f32(false, av, false, bv,
                                                 (short)0, acc2, false, false);
  }
  {
    const float bias = bd2[n];
#pragma unroll
    for (int i = 0; i < 8; ++i)
      out[(rowBase + i + half * 8) * 16 + n] = acc2[i] + bias;
  }
}

// ---------------------------------------------------------------------------
extern "C" void kernel_launch(void* const* d_in, const int* in_sizes, int n_in,
                              void* d_out, int out_size, void* d_ws, size_t ws_size,
                              hipStream_t stream) {
  const float* x   = (const float*)d_in[0];
  const float* We1 = (const float*)d_in[1];
  const float* be1 = (const float*)d_in[2];
  const float* We2 = (const float*)d_in[3];
  const float* be2 = (const float*)d_in[4];
  const float* Wd1 = (const float*)d_in[5];
  const float* bd1 = (const float*)d_in[6];
  const float* Wd2 = (const float*)d_in[7];
  const float* bd2 = (const float*)d_in[8];
  const float* hlp = (const float*)d_in[9];   // h_lowpass
  const float* hps = (const float*)d_in[10];  // h_pulseshape
  const float* noise = (const float*)d_in[11];
  const int*   snr   = (const int*)d_in[12];
  float* ws = (float*)d_ws;

  const int nrows = in_sizes[0] / 16;         // B = 131072
  const int grid  = (nrows + 127) / 128;      // 8 waves/block, 16 rows/wave

  tdm_filt_kernel<<<7, 160, 0, stream>>>(hps, hlp, ws);
  enc_kernel<<<grid, 256, 0, stream>>>(x, We1, be1, We2, be2, ws, nrows);
  dec_kernel<<<grid, 256, 0, stream>>>(ws, Wd1, bd1, Wd2, bd2, noise, snr,
                                       (float*)d_out, nrows);
}